// BoxDecoder_55104430407865
// MI455X (gfx1250) — compile-verified
//
#include <hip/hip_runtime.h>
#include <stdint.h>

// ---------------------------------------------------------------------------
// BoxDecoder (Gumbel soft-box containment probs), MI455X / gfx1250.
//
//   out[i,j] = exp( min( interVol(i,j) - vol1(i), 0 ) ),  64 x 10331, D=128
//
// Transcendental-bound. All math in base-2 on raw hw ops (v_exp_f32 = 2^x,
// v_log_f32 = log2, v_rcp_f32), args provably in safe range (no fixups, no
// denorms; EPS clamp never binds since softplus(t)/tau >= 0.2).
// Trans reduction tricks:
//   * exp(softplus(o)) == 1 + e^o exactly  -> corners need 2 exp + 3 rcp
//   * sum_k log2(l1_k) == sum_grp log2(prod_8 l1_k), l1 in (0.36,2.17) so an
//     8-product stays in [3e-4, 483] -> outer log2 amortized 8x
//   * per-element constant ln(ln2/tau) cancels inter vs vol1:
//     out = 2^min(S2_int - S2_vol, 0)  (single v_exp_f32 at the end)
//
// WMMA does not apply: the (i,j,k) term is min/max + log-softplus, not
// factorable as A[i,k]*B[k,j]. CDNA5 paths: TDM tensor_load_to_lds (3D tile
// descriptor, TENSORcnt) stages k-major mention boxes into LDS for b128 DS
// reads; s_wait_tensorcnt; global_prefetch_b8; wave32 blocking.
// ---------------------------------------------------------------------------

#define NM   64      // mentions
#define NT   10331   // types
#define BD   128     // box dim
#define MPER 16      // mentions per block (pair kernel)
#define JT   256     // types per block   (pair kernel)
#define KG   8       // k-group size for product-of-l1 trick

constexpr float INVT  = 1.2471085395024732f;   // INV_SOFTPLUS_TEMP
constexpr float C0    = 2.0f * 0.5772156649015329f * 0.00036026463511690845f;
constexpr float LOG2E = 1.4426950408889634f;
constexpr float K1    = INVT * LOG2E;          // d-scale into base-2 exp arg
constexpr float K0    = -C0 * INVT * LOG2E;    // folded -2*gamma*beta offset

typedef unsigned int u32x4 __attribute__((ext_vector_type(4)));
typedef int          i32x8 __attribute__((ext_vector_type(8)));
typedef int          i32x4 __attribute__((ext_vector_type(4)));

__device__ __forceinline__ float exp2_hw(float x) { return __builtin_amdgcn_exp2f(x); } // v_exp_f32
__device__ __forceinline__ float log2_hw(float x) { return __builtin_amdgcn_logf(x); }  // v_log_f32
__device__ __forceinline__ float rcp_hw(float x)  { return __builtin_amdgcn_rcpf(x); }  // v_rcp_f32

// l1 = log2(1 + 2^t2), t2 = (d - 2*gamma*beta)*invT*log2e   (2 trans, 2 VALU)
__device__ __forceinline__ float l1_term(float d) {
    float t2 = fmaf(d, K1, K0);
    return log2_hw(1.0f + exp2_hw(t2));
}

// box corners from (c, o):  es = exp(softplus(o)) = 1 + e^o (exact identity)
//   z = sigmoid(c - sp) = rcp(1 + e^-c * es)
//   Z = sigmoid(c + sp) = rcp(1 + e^-c / es)
__device__ __forceinline__ void corners_hw(float c, float o, float& z, float& Z) {
    const float ec = exp2_hw(-c * LOG2E);
    const float es = 1.0f + exp2_hw(o * LOG2E);
    z = rcp_hw(fmaf(ec, es, 1.0f));
    Z = rcp_hw(fmaf(ec, rcp_hw(es), 1.0f));
}

// -------------------------------------------------------------------------
// Kernel 1: mention boxes, k-major:  z1 -> ws[k*64+i] (k=0..127, i=0..63),
//           Z1 -> ws[8192 + k*64+i]; self-volume base-2 sums -> ws[16384+i]
// -------------------------------------------------------------------------
__global__ __launch_bounds__(BD)
void box_mentions_kernel(const float* __restrict__ me, float* __restrict__ ws) {
    const int i = blockIdx.x;    // mention
    const int k = threadIdx.x;   // dim
    const float c = me[i * 2 * BD + k];
    const float o = me[i * 2 * BD + BD + k];
    float z, Z;
    corners_hw(c, o, z, Z);
    ws[k * NM + i]           = z;   // k-major for b128 LDS reads in pair kernel
    ws[NM * BD + k * NM + i] = Z;

    float lv = log2_hw(l1_term(Z - z));
    __shared__ float red[BD];
    red[k] = lv;
    __syncthreads();
    #pragma unroll
    for (int s = BD / 2; s > 0; s >>= 1) {
        if (k < s) red[k] += red[k + s];
        __syncthreads();
    }
    if (k == 0) ws[2 * NM * BD + i] = red[0];
}

// -------------------------------------------------------------------------
// Kernel 2: 256 types x 16 mentions per block; TDM 3D-tile stages the k-major
// mention slice into LDS; 8-wide k-groups accumulate products of l1.
// -------------------------------------------------------------------------
__global__ __launch_bounds__(JT)
void pair_kernel(const float* __restrict__ te, const float* __restrict__ ws,
                 float* __restrict__ out) {
    __shared__ float lds_zZ[2 * BD * MPER];   // 16 KB: [z|Z][k][i], rows of 16

    const int tid = threadIdx.x;
    const int j   = blockIdx.x * JT + tid;
    const int jc  = min(j, NT - 1);           // clamp so EXEC stays full
    const int ib  = blockIdx.y * MPER;

    // ---- TDM 3D tile: x=16 mentions @ offset ib, y=128 dims (stride 64),
    //      z=2 regions (z1 then Z1, stride 8192 floats) -> LDS linear ----
    if (tid < 32) {   // wave 0 issues the DMA (TDM is per-wave, EXEC-ignored)
        const uint64_t gaddr = (uint64_t)(uintptr_t)(ws + ib);
        const uint32_t laddr = (uint32_t)(uintptr_t)(&lds_zZ[0]); // LDS offset = low 32b
        u32x4 g0;
        g0[0] = 1u;                                             // count=1, user descriptor
        g0[1] = laddr;                                          // lds_addr
        g0[2] = (uint32_t)(gaddr & 0xFFFFFFFFull);              // global_addr[31:0]
        g0[3] = (uint32_t)((gaddr >> 32) & 0x01FFFFFFull)       // global_addr[56:32]
              | (2u << 30);                                     // type=2 ("image")
        i32x8 g1;
        g1[0] = (int)(2u << 16);                     // data_size=2 (4B); wg_mask=0
        g1[1] = (int)((uint32_t)NM << 16);           // tensor_dim0 = 64
        g1[2] = (int)((uint32_t)BD << 16);           // tensor_dim1 = 128
        g1[3] = (int)((uint32_t)MPER << 16);         // tile_dim0 = 16
        g1[4] = (int)(BD | (2u << 16));              // tile_dim1 = 128, tile_dim2 = 2
        g1[5] = NM;                                  // tensor_dim0_stride = 64
        g1[6] = (int)((uint32_t)(NM * BD) << 16);    // tensor_dim1_stride = 8192 (lo16)
        g1[7] = (int)((uint32_t)(NM * BD) >> 16);    // tensor_dim1_stride (hi bits) = 0
        i32x4 g2 = {2, 0, 0, 0};                     // tensor_dim2 = 2; tile_dim3 = 0
        i32x4 g3 = {0, 0, 0, 0};
#if __has_include(<hip/amd_detail/amd_gfx1250_TDM.h>)
        i32x8 g4 = {0, 0, 0, 0, 0, 0, 0, 0};         // therock headers: 6-arg builtin
        __builtin_amdgcn_tensor_load_to_lds(g0, g1, g2, g3, g4, 0);
#else
        __builtin_amdgcn_tensor_load_to_lds(g0, g1, g2, g3, 0);  // ROCm 7.2: 5-arg
#endif
        __builtin_amdgcn_s_wait_tensorcnt(0);
    }
    __syncthreads();

    const float* __restrict__ rowc = te + (size_t)jc * (2 * BD);

    float acc[MPER];
    #pragma unroll
    for (int i = 0; i < MPER; ++i) acc[i] = 0.0f;

    for (int kk = 0; kk < BD; kk += KG) {
        const float4 c4a = *(const float4*)(rowc + kk);
        const float4 c4b = *(const float4*)(rowc + kk + 4);
        const float4 o4a = *(const float4*)(rowc + BD + kk);
        const float4 o4b = *(const float4*)(rowc + BD + kk + 4);
        __builtin_prefetch(rowc + kk + 32, 0, 0);   // global_prefetch_b8
        const float cv[KG] = {c4a.x, c4a.y, c4a.z, c4a.w, c4b.x, c4b.y, c4b.z, c4b.w};
        const float ov[KG] = {o4a.x, o4a.y, o4a.z, o4a.w, o4b.x, o4b.y, o4b.z, o4b.w};

        float prod[MPER];
        #pragma unroll
        for (int i = 0; i < MPER; ++i) prod[i] = 1.0f;

        #pragma unroll
        for (int u = 0; u < KG; ++u) {
            const int k = kk + u;
            float zb, Zb;                             // type box corners for dim k
            corners_hw(cv[u], ov[u], zb, Zb);
            // 16 mention values for dim k: 4+4 ds_load_b128 (broadcast)
            const float4* zrow = (const float4*)(lds_zZ + k * MPER);
            const float4* Zrow = (const float4*)(lds_zZ + BD * MPER + k * MPER);
            const float4 za = zrow[0], zc = zrow[1], ze = zrow[2], zg = zrow[3];
            const float4 Za = Zrow[0], Zc = Zrow[1], Ze = Zrow[2], Zg = Zrow[3];
            const float zi[MPER] = {za.x, za.y, za.z, za.w, zc.x, zc.y, zc.z, zc.w,
                                    ze.x, ze.y, ze.z, ze.w, zg.x, zg.y, zg.z, zg.w};
            const float Zi[MPER] = {Za.x, Za.y, Za.z, Za.w, Zc.x, Zc.y, Zc.z, Zc.w,
                                    Ze.x, Ze.y, Ze.z, Ze.w, Zg.x, Zg.y, Zg.z, Zg.w};
            #pragma unroll
            for (int i = 0; i < MPER; ++i) {
                const float a = fmaxf(zi[i], zb);     // min_pt
                const float b = fminf(Zi[i], Zb);     // max_pt
                prod[i] *= l1_term(b - a);            // 2 trans + 4 VALU
            }
        }
        #pragma unroll
        for (int i = 0; i < MPER; ++i) acc[i] += log2_hw(prod[i]); // amortized log
    }

    if (j < NT) {
        const float* __restrict__ volg = ws + 2 * NM * BD;
        #pragma unroll
        for (int i = 0; i < MPER; ++i) {
            // out = exp(min(inter - vol1, 0)) = 2^min(S2_int - S2_vol, 0)
            const float lp2 = fminf(acc[i] - volg[ib + i], 0.0f);
            out[(size_t)(ib + i) * NT + j] = exp2_hw(lp2);
        }
    }
}

// -------------------------------------------------------------------------
extern "C" void kernel_launch(void* const* d_in, const int* in_sizes, int n_in,
                              void* d_out, int out_size, void* d_ws, size_t ws_size,
                              hipStream_t stream) {
    (void)in_sizes; (void)n_in; (void)out_size; (void)ws_size;
    const float* mention_emb = (const float*)d_in[0];   // [64, 256]
    const float* type_emb    = (const float*)d_in[1];   // [10331, 256]
    float* out = (float*)d_out;                         // [64, 10331]
    float* ws  = (float*)d_ws;                          // needs 16448 floats = 65.8 KB

    box_mentions_kernel<<<dim3(NM), dim3(BD), 0, stream>>>(mention_emb, ws);

    dim3 grid((NT + JT - 1) / JT, NM / MPER);           // 41 x 4 blocks
    pair_kernel<<<grid, dim3(JT), 0, stream>>>(type_emb, ws, out);
}